// RoutingFreeGate_15702400434203
// MI455X (gfx1250) — compile-verified
//
#include <hip/hip_runtime.h>
#include <math.h>
#include <stdint.h>

// gate_hidden = (mask?x:0)[N,H] @ W_A^T[H,R]; score = ||row||_2*scale - bias;
// gmask = mask & (score >= 0.5); outputs concat: [mask(N), score(N), hidden(N*R)].
// N=16384, H=2048, R=512 (derived from in_sizes, divisibility assumed).
//
// FP32 GEMM on the matrix pipe via V_WMMA_F32_16X16X4_F32.
// Tiles stream into LDS with CDNA5 async DMA (global_load_async_to_lds_b128,
// ASYNCcnt), double-buffered; mask is applied at the epilogue (hidden rows of
// invalid tokens are identically zero either way).

typedef float v2f __attribute__((ext_vector_type(2)));
typedef float v8f __attribute__((ext_vector_type(8)));

constexpr int BM = 128;           // rows (tokens) per block
constexpr int BN = 128;           // cols (R) per block
constexpr int BK = 32;            // K chunk
constexpr int PITCH = BK + 4;     // 36 dwords: 16B-aligned rows, conflict-free b64 reads

__global__ __launch_bounds__(256, 2)
void gate_gemm_kernel(const float* __restrict__ x,
                      const unsigned char* __restrict__ mask,  // jnp bool = 1 byte
                      const float* __restrict__ W,             // [R,H] row-major
                      float* __restrict__ outHidden,           // [N,R]
                      float* __restrict__ wsSumSq,             // [N] (pre-zeroed)
                      int H, int R)
{
    __shared__ float sX[2][BM * PITCH];
    __shared__ float sW[2][BN * PITCH];

    const int tid     = threadIdx.x;
    const int rowBase = blockIdx.y * BM;
    const int colBase = blockIdx.x * BN;

    // ---- async DMA staging: 128x32 tile = 1024 float4; 256 threads -> 4 each ----
    const int srow = tid >> 3;            // 0..31 (+ i*32)
    const int scol = (tid & 7) << 2;      // 0,4,...,28

    auto issueAsync = [&](int kc, int buf) {
        const int k0 = kc * BK;
        #pragma unroll
        for (int i = 0; i < 4; ++i) {
            const int row = i * 32 + srow;
            const uint32_t ldsX = (uint32_t)(uintptr_t)&sX[buf][row * PITCH + scol];
            const uint64_t gx   = (uint64_t)(uintptr_t)(x + (size_t)(rowBase + row) * H + k0 + scol);
            asm volatile("global_load_async_to_lds_b128 %0, %1, off"
                         :: "v"(ldsX), "v"(gx) : "memory");
            const uint32_t ldsW = (uint32_t)(uintptr_t)&sW[buf][row * PITCH + scol];
            const uint64_t gw   = (uint64_t)(uintptr_t)(W + (size_t)(colBase + row) * H + k0 + scol);
            asm volatile("global_load_async_to_lds_b128 %0, %1, off"
                         :: "v"(ldsW), "v"(gw) : "memory");
        }
    };

    // ---- wave tiling: 8 waves as 2(M) x 4(N); wave tile 64x32 = 4x2 WMMA tiles ----
    const int lane = tid & 31;
    const int wid  = tid >> 5;
    const int wm   = (wid & 1) * 64;
    const int wn   = (wid >> 1) * 32;
    const int lr   = lane & 15;
    const int kho  = (lane >> 4) << 1;   // lo half: K=k..k+1, hi half: K=k+2..k+3

    v8f acc[4][2];
    #pragma unroll
    for (int mi = 0; mi < 4; ++mi)
        #pragma unroll
        for (int ni = 0; ni < 2; ++ni)
            acc[mi][ni] = (v8f){0.f, 0.f, 0.f, 0.f, 0.f, 0.f, 0.f, 0.f};

    auto compute = [&](int buf) {
        const float* sx = sX[buf];
        const float* sw = sW[buf];
        #pragma unroll
        for (int k = 0; k < BK; k += 4) {
            v2f a[4], b[2];
            #pragma unroll
            for (int mi = 0; mi < 4; ++mi)
                a[mi] = *(const v2f*)&sx[(wm + mi * 16 + lr) * PITCH + k + kho];
            #pragma unroll
            for (int ni = 0; ni < 2; ++ni)
                b[ni] = *(const v2f*)&sw[(wn + ni * 16 + lr) * PITCH + k + kho];
            #pragma unroll
            for (int mi = 0; mi < 4; ++mi)
                #pragma unroll
                for (int ni = 0; ni < 2; ++ni)
                    acc[mi][ni] = __builtin_amdgcn_wmma_f32_16x16x4_f32(
                        false, a[mi], false, b[ni], (short)0, acc[mi][ni], false, false);
        }
    };

    // ---- main loop: async DMA fills buf^1 while WMMAs consume buf ----
    const int nk = H / BK;
    issueAsync(0, 0);
    int cur = 0;
    for (int kc = 0; kc < nk; ++kc) {
        asm volatile("s_wait_asynccnt 0" ::: "memory");  // our DMA into buf[cur] done
        __syncthreads();                                  // everyone's DMA visible
        if (kc + 1 < nk) issueAsync(kc + 1, cur ^ 1);     // in flight during compute
        compute(cur);
        cur ^= 1;
    }

    // ---- epilogue: apply mask, store hidden tile, per-row sum of squares ----
    // C/D layout: vgpr v, lanes 0-15 -> row base+v col=lr ; lanes 16-31 -> row base+8+v
    const size_t obase = (size_t)rowBase * R + colBase;
    #pragma unroll
    for (int mi = 0; mi < 4; ++mi) {
        const int rhalf = wm + mi * 16 + ((lane >> 4) << 3);
        #pragma unroll
        for (int v = 0; v < 8; ++v) {
            const int r = rhalf + v;
            const float mval = (mask[rowBase + r] != 0) ? 1.0f : 0.0f;
            float ss = 0.f;
            #pragma unroll
            for (int ni = 0; ni < 2; ++ni) {
                const float val = acc[mi][ni][v] * mval;
                outHidden[obase + (size_t)r * R + wn + ni * 16 + lr] = val;
                ss = fmaf(val, val, ss);
            }
            // reduce across the 16 lanes holding this row (xor<16 stays in half)
            ss += __shfl_xor(ss, 8, 32);
            ss += __shfl_xor(ss, 4, 32);
            ss += __shfl_xor(ss, 2, 32);
            ss += __shfl_xor(ss, 1, 32);
            if (lr == 0)
                atomicAdd(&wsSumSq[rowBase + r], ss);
        }
    }
}

__global__ __launch_bounds__(256)
void gate_finalize_kernel(const float* __restrict__ ss,
                          const unsigned char* __restrict__ mask,
                          const float* __restrict__ scale,
                          const float* __restrict__ bias,
                          float* __restrict__ outMask,
                          float* __restrict__ outScore,
                          int N)
{
    const int i = blockIdx.x * blockDim.x + threadIdx.x;
    if (i >= N) return;
    const float score = sqrtf(ss[i]) * scale[0] - bias[0];
    const bool g = (mask[i] != 0) && (score >= 0.5f);   // GATE_THRESHOLD / GATE_TEMPERATURE
    outMask[i]  = g ? 1.0f : 0.0f;
    outScore[i] = g ? score : -__builtin_inff();
}

extern "C" void kernel_launch(void* const* d_in, const int* in_sizes, int n_in,
                              void* d_out, int out_size, void* d_ws, size_t ws_size,
                              hipStream_t stream)
{
    const float*         x     = (const float*)d_in[0];
    const unsigned char* mask  = (const unsigned char*)d_in[1];
    const float*         W     = (const float*)d_in[2];
    const float*         scale = (const float*)d_in[3];
    const float*         bias  = (const float*)d_in[4];

    const int N = in_sizes[1];              // B*T tokens
    const int H = in_sizes[0] / N;          // hidden
    const int R = in_sizes[2] / H;          // rank

    float* out       = (float*)d_out;
    float* outMask   = out;
    float* outScore  = out + N;
    float* outHidden = out + 2 * (size_t)N;
    float* ws        = (float*)d_ws;

    // d_ws is not re-poisoned between replays; zero it every call (graph-capturable).
    hipMemsetAsync(ws, 0, (size_t)N * sizeof(float), stream);

    dim3 grid(R / BN, N / BM);
    gate_gemm_kernel<<<grid, 256, 0, stream>>>(x, mask, W, outHidden, ws, H, R);

    gate_finalize_kernel<<<(N + 255) / 256, 256, 0, stream>>>(
        ws, mask, scale, bias, outMask, outScore, N);
}